// FedGNN_LGCN_Encoder_42228118454598
// MI455X (gfx1250) — compile-verified
//
#include <hip/hip_runtime.h>

typedef __attribute__((ext_vector_type(2))) float v2f;
typedef __attribute__((ext_vector_type(8))) float v8f;

#define DIMS 64
#define TPB  256
#define PADREC 8   // sentinel records past the end of the sorted edge list

// ---------------------------------------------------------------------------
// Init: ego0 = concat(user_emb, item_emb); counts/cursor = 0
// (acc/d_out needs no init: layer-1 epilogue overwrites it fully)
// ---------------------------------------------------------------------------
__global__ void k_init(const float* __restrict__ ue, const float* __restrict__ ie,
                       float* __restrict__ ego,
                       int* __restrict__ counts, int* __restrict__ cursor,
                       int uElems, int total, int nNodes) {
  int i = blockIdx.x * blockDim.x + threadIdx.x;
  if (i < total) ego[i] = (i < uElems) ? ue[i] : ie[i - uElems];
  if (i < nNodes) { counts[i] = 0; cursor[i] = 0; }
}

// ---------------------------------------------------------------------------
// CSR build pass 1: per-destination degree histogram
// ---------------------------------------------------------------------------
__global__ void k_count(const int* __restrict__ dst, int* __restrict__ counts, int nE) {
  int e = blockIdx.x * blockDim.x + threadIdx.x;
  if (e < nE) atomicAdd(&counts[dst[e]], 1);
}

// ---------------------------------------------------------------------------
// CSR build pass 2: exclusive scan over N counts (single workgroup, chunked)
// ---------------------------------------------------------------------------
__global__ void __launch_bounds__(1024)
k_scan(const int* __restrict__ counts, int* __restrict__ row_start, int nNodes) {
  __shared__ int ls[1024];
  int t = threadIdx.x;
  int chunk = (nNodes + 1023) / 1024;
  int b = t * chunk;
  int e = min(b + chunk, nNodes);
  int s = 0;
  for (int i = b; i < e; ++i) s += counts[i];
  ls[t] = s;
  __syncthreads();
  for (int off = 1; off < 1024; off <<= 1) {
    int v = (t >= off) ? ls[t - off] : 0;
    __syncthreads();
    ls[t] += v;
    __syncthreads();
  }
  int p = ls[t] - s;  // exclusive prefix of this thread's chunk
  for (int i = b; i < e; ++i) { row_start[i] = p; p += counts[i]; }
  if (t == 0) row_start[nNodes] = ls[1023];
}

// ---------------------------------------------------------------------------
// CSR build pass 3: scatter edges into dst-sorted SoA arrays {src, dst, val}
// plus sentinel pads (dst=-1 never matches any row -> A entry = 0)
// ---------------------------------------------------------------------------
__global__ void k_scatter(const int* __restrict__ src, const int* __restrict__ dst,
                          const float* __restrict__ vals,
                          const int* __restrict__ row_start, int* __restrict__ cursor,
                          int* __restrict__ sSrc, int* __restrict__ sDst,
                          float* __restrict__ sVal, int nE) {
  int e = blockIdx.x * blockDim.x + threadIdx.x;
  if (e < nE) {
    int d = dst[e];
    int pos = row_start[d] + atomicAdd(&cursor[d], 1);
    sSrc[pos] = src[e];
    sDst[pos] = d;
    sVal[pos] = vals[e];
  }
  if (e < PADREC) {
    sSrc[nE + e] = 0;
    sDst[nE + e] = -1;
    sVal[nE + e] = 0.0f;
  }
}

// ---------------------------------------------------------------------------
// SpMM layer on the matrix pipe. One block = 16 destination rows.
// Wave w handles dim tile [16w, 16w+16). Per trip: 8 sorted edges -> 2x
//   A[16x4] = one-hot(dst)*val, B[4x16] = gathered ego slices,
//   C += A*B via v_wmma_f32_16x16x4_f32. Branch-free inner loop:
//   out-of-group records self-mask (dst mismatch => A=0).
// Records stream through the scalar path (contiguous SoA -> wide s_load);
// ego gathers use 32-bit offsets (N*D < 2^31) -> saddr-form global loads.
// Epilogue is compile-time specialized per layer:
//   FIRST=1 : acc = c (no acc read)         (layer 1)
//   WEGO=1  : also store layer output to eout (all but last layer)
// ---------------------------------------------------------------------------
template <int FIRST, int WEGO>
__global__ void __launch_bounds__(128)
k_spmm(const int* __restrict__ sSrc, const int* __restrict__ sDst,
       const float* __restrict__ sVal, const int* __restrict__ row_start,
       const float* __restrict__ ein, float* __restrict__ eout,
       float* __restrict__ acc, int nNodes) {
  int g    = blockIdx.x;
  int tile = threadIdx.x >> 5;   // 0..3 : which 16-dim column tile
  int lane = threadIdx.x & 31;
  int half = lane >> 4;          // K-group: lanes 0-15 -> K={0,1}, 16-31 -> K={2,3}
  int mn   = lane & 15;          // row (A) / column (B,C,D) index
  int r0   = g * 16;
  int rHi  = min(r0 + 16, nNodes);
  // wave-uniform edge range -> scalar loop bounds, scalar record loads
  int eBeg = __builtin_amdgcn_readfirstlane(row_start[r0]);
  int eEnd = __builtin_amdgcn_readfirstlane(row_start[rHi]);
  int dim   = tile * 16 + mn;
  int myrow = r0 + mn;

  v8f c = {0.f, 0.f, 0.f, 0.f, 0.f, 0.f, 0.f, 0.f};

  for (int kk = eBeg; kk < eEnd; kk += 8) {
    // 8 wave-uniform records via 3 contiguous 32B scalar runs
    const int*   sp = sSrc + kk;
    const int*   dp = sDst + kk;
    const float* vp = sVal + kk;
    int  x0 = sp[0], x1 = sp[1], x2 = sp[2], x3 = sp[3];
    int  x4 = sp[4], x5 = sp[5], x6 = sp[6], x7 = sp[7];
    int  d0 = dp[0], d1 = dp[1], d2 = dp[2], d3 = dp[3];
    int  d4 = dp[4], d5 = dp[5], d6 = dp[6], d7 = dp[7];
    float w0 = vp[0], w1 = vp[1], w2 = vp[2], w3 = vp[3];
    float w4 = vp[4], w5 = vp[5], w6 = vp[6], w7 = vp[7];

    // lane-half selects its K slots: {0,1} or {2,3} within each 4-edge batch
    int   s0x = half ? x2 : x0;  int   s0y = half ? d2 : d0;  float s0w = half ? w2 : w0;
    int   s1x = half ? x3 : x1;  int   s1y = half ? d3 : d1;  float s1w = half ? w3 : w1;
    int   s2x = half ? x6 : x4;  int   s2y = half ? d6 : d4;  float s2w = half ? w6 : w4;
    int   s3x = half ? x7 : x5;  int   s3y = half ? d7 : d5;  float s3w = half ? w7 : w5;

    // coalesced ego gathers: 32-bit element offsets (saddr + voffset form)
    float g0 = ein[(unsigned)(s0x * DIMS + dim)];
    float g1 = ein[(unsigned)(s1x * DIMS + dim)];
    float g2 = ein[(unsigned)(s2x * DIMS + dim)];
    float g3 = ein[(unsigned)(s3x * DIMS + dim)];

    v2f a0, b0, a1, b1;
    a0[0] = (s0y == myrow) ? s0w : 0.0f;
    a0[1] = (s1y == myrow) ? s1w : 0.0f;
    a1[0] = (s2y == myrow) ? s2w : 0.0f;
    a1[1] = (s3y == myrow) ? s3w : 0.0f;
    b0[0] = g0;  b0[1] = g1;
    b1[0] = g2;  b1[1] = g3;

    c = __builtin_amdgcn_wmma_f32_16x16x4_f32(false, a0, false, b0, (short)0, c,
                                              false, false);
    c = __builtin_amdgcn_wmma_f32_16x16x4_f32(false, a1, false, b1, (short)0, c,
                                              false, false);
  }

  // C/D layout: VGPR v, lane-half h -> row M = v + 8h; column N = mn
  unsigned base = (unsigned)((r0 + 8 * half) * DIMS + dim);
  if (r0 + 16 <= nNodes) {            // uniform: full tile, no per-element guard
#pragma unroll
    for (int v = 0; v < 8; ++v) {
      unsigned idx = base + (unsigned)(v * DIMS);
      if (FIRST) acc[idx] = c[v];     // overwrite: no acc read on layer 1
      else       acc[idx] += c[v];
      if (WEGO)  eout[idx] = c[v];
    }
  } else {                            // ragged last group (N % 16 != 0 only)
#pragma unroll
    for (int v = 0; v < 8; ++v) {
      int row = r0 + v + 8 * half;
      if (row < nNodes) {
        unsigned idx = (unsigned)(row * DIMS + dim);
        if (FIRST) acc[idx] = c[v]; else acc[idx] += c[v];
        if (WEGO)  eout[idx] = c[v];
      }
    }
  }
}

__global__ void k_scale(float* __restrict__ out, float s, int n) {
  int i = blockIdx.x * blockDim.x + threadIdx.x;
  if (i < n) out[i] *= s;
}

// ---------------------------------------------------------------------------
static inline size_t align256(size_t x) { return (x + 255) & ~(size_t)255; }

extern "C" void kernel_launch(void* const* d_in, const int* in_sizes, int n_in,
                              void* d_out, int out_size, void* d_ws, size_t ws_size,
                              hipStream_t stream) {
  const float* user_emb = (const float*)d_in[0];
  const float* item_emb = (const float*)d_in[1];
  const int*   edge_src = (const int*)d_in[2];
  const int*   edge_dst = (const int*)d_in[3];
  const float* edge_val = (const float*)d_in[4];

  const int uElems = in_sizes[0];            // U*64
  const int iElems = in_sizes[1];            // I*64
  const int nE     = in_sizes[2];            // 2E
  const int U      = uElems / DIMS;
  const int I      = iElems / DIMS;
  const int N      = U + I;
  const int total  = N * DIMS;
  const int L      = 3;

  // Workspace carve-up (~95 MB): two ego ping-pong buffers, SoA records, CSR
  char* p = (char*)d_ws;
  float* ego_a  = (float*)p;  p += align256((size_t)total * sizeof(float));
  float* ego_b  = (float*)p;  p += align256((size_t)total * sizeof(float));
  int*   sSrc   = (int*)p;    p += align256((size_t)(nE + PADREC) * sizeof(int));
  int*   sDst   = (int*)p;    p += align256((size_t)(nE + PADREC) * sizeof(int));
  float* sVal   = (float*)p;  p += align256((size_t)(nE + PADREC) * sizeof(float));
  int*   rowptr = (int*)p;    p += align256((size_t)(N + 1) * sizeof(int));
  int*   counts = (int*)p;    p += align256((size_t)N * sizeof(int));
  int*   cursor = (int*)p;    p += align256((size_t)N * sizeof(int));

  float* acc = (float*)d_out;

  int gInit = (total + TPB - 1) / TPB;
  k_init<<<gInit, TPB, 0, stream>>>(user_emb, item_emb, ego_a,
                                    counts, cursor, uElems, total, N);

  int gE = (nE + TPB - 1) / TPB;
  k_count<<<gE, TPB, 0, stream>>>(edge_dst, counts, nE);
  k_scan<<<1, 1024, 0, stream>>>(counts, rowptr, N);
  k_scatter<<<gE, TPB, 0, stream>>>(edge_src, edge_dst, edge_val,
                                    rowptr, cursor, sSrc, sDst, sVal, nE);

  int nGroups = (N + 15) / 16;
  // Layer 1: acc = A@ego0, ego1 = A@ego0
  k_spmm<1, 1><<<nGroups, 128, 0, stream>>>(sSrc, sDst, sVal, rowptr,
                                            ego_a, ego_b, acc, N);
  // Layer 2: acc += A@ego1, ego2 = A@ego1
  k_spmm<0, 1><<<nGroups, 128, 0, stream>>>(sSrc, sDst, sVal, rowptr,
                                            ego_b, ego_a, acc, N);
  // Layer 3: acc += A@ego2 (layer output consumed only through acc)
  k_spmm<0, 0><<<nGroups, 128, 0, stream>>>(sSrc, sDst, sVal, rowptr,
                                            ego_a, ego_b, acc, N);

  k_scale<<<gInit, TPB, 0, stream>>>(acc, 1.0f / (float)L, total);
}